// EAEF_2138893713413
// MI455X (gfx1250) — compile-verified
//
#include <hip/hip_runtime.h>
#include <math.h>

// Problem constants (B,C,H,W) = (4,128,160,160)
#define Bb   4
#define Cc   128
#define C2c  256
#define Hh   160
#define Ww   160
#define HWp  (Hh*Ww)          // 25600
#define CAH  16               // CA_HID

typedef float v2f __attribute__((ext_vector_type(2)));
typedef float v8f __attribute__((ext_vector_type(8)));

__device__ __forceinline__ float gelu_exact(float x) {
    return 0.5f * x * (1.0f + erff(x * 0.70710678118654752440f));
}
__device__ __forceinline__ float sigmoidf_(float x) {
    return 1.0f / (1.0f + __expf(-x));
}

// D = A(16xK) * B(KxN tile) + D via V_WMMA_F32_16X16X4_F32.
// Lane layout (ISA 7.12.2, 32-bit operands, wave32):
//   A 16x4:  a.x = A[M=lane&15][k0 + (lane>=16?2:0)], a.y = same +1
//   B 4x16:  b.x = B[k0 + (lane>=16?2:0)][N=lane&15], b.y = same +1
//   C 16x16: c[r] = C[M = r + (lane>=16?8:0)][N = lane&15]
// ---------------------------------------------------------------------------

// ---- Kernel 1: spatial mean of RGB and T -> stacked 16x128 (rows 0-3 RGB, 4-7 T)
__global__ __launch_bounds__(256) void k_pool_mean(const float* __restrict__ RGB,
                                                   const float* __restrict__ T,
                                                   float* __restrict__ Mstack) {
    int blk = blockIdx.x;                 // 0..1023
    int tensor = blk >> 9;                // 0=RGB, 1=T
    int bc = blk & 511;
    int b = bc >> 7, c = bc & 127;
    const float* src = (tensor ? T : RGB) + (size_t)(b * Cc + c) * HWp;
    const float4* s4 = (const float4*)src;
    float sum = 0.f;
    for (int i = threadIdx.x; i < HWp / 4; i += 256) {
        float4 v = s4[i];
        sum += (v.x + v.y) + (v.z + v.w);
    }
    __shared__ float red[256];
    red[threadIdx.x] = sum;
    __syncthreads();
    for (int s = 128; s > 0; s >>= 1) {
        if (threadIdx.x < s) red[threadIdx.x] += red[threadIdx.x + s];
        __syncthreads();
    }
    if (threadIdx.x == 0)
        Mstack[(tensor * 4 + b) * Cc + c] = red[0] * (1.0f / (float)HWp);
}

// ---- Kernel 2: H1(16x256) = gelu( Mstack(16x128) @ fp_down_w^T + b ), WMMA
__global__ __launch_bounds__(512) void k_fp_gemm1(const float* __restrict__ M,
                                                  const float* __restrict__ W,   // (256,128)
                                                  const float* __restrict__ bias,
                                                  float* __restrict__ H1) {
    const int lane = threadIdx.x & 31;
    const int wave = threadIdx.x >> 5;    // 0..15 -> N tile
    const int n0 = wave * 16;
    const int mr = lane & 15;
    const int kh = (lane >> 4) * 2;
    const int n = n0 + mr;
    v8f acc = {};
    for (int k0 = 0; k0 < 128; k0 += 4) {
        v2f a, bm;
        a.x  = M[mr * 128 + k0 + kh];
        a.y  = M[mr * 128 + k0 + kh + 1];
        bm.x = W[n * 128 + k0 + kh];       // B[k][n] = W^T[k][n] = W[n][k]
        bm.y = W[n * 128 + k0 + kh + 1];
        acc = __builtin_amdgcn_wmma_f32_16x16x4_f32(false, a, false, bm,
                                                    (short)0, acc, false, false);
    }
    float bn = bias[n];
    int mbase = (lane >> 4) * 8;
#pragma unroll
    for (int r = 0; r < 8; ++r)
        H1[(mbase + r) * 256 + n] = gelu_exact(acc[r] + bn);
}

// ---- Kernel 3: Y(16x128) = H1(16x256) @ fp_up_w^T + b, WMMA
__global__ __launch_bounds__(256) void k_fp_gemm2(const float* __restrict__ H1,
                                                  const float* __restrict__ W,   // (128,256)
                                                  const float* __restrict__ bias,
                                                  float* __restrict__ Y) {
    const int lane = threadIdx.x & 31;
    const int wave = threadIdx.x >> 5;    // 0..7
    const int n0 = wave * 16;
    const int mr = lane & 15;
    const int kh = (lane >> 4) * 2;
    const int n = n0 + mr;
    v8f acc = {};
    for (int k0 = 0; k0 < 256; k0 += 4) {
        v2f a, bm;
        a.x  = H1[mr * 256 + k0 + kh];
        a.y  = H1[mr * 256 + k0 + kh + 1];
        bm.x = W[n * 256 + k0 + kh];
        bm.y = W[n * 256 + k0 + kh + 1];
        acc = __builtin_amdgcn_wmma_f32_16x16x4_f32(false, a, false, bm,
                                                    (short)0, acc, false, false);
    }
    float bn = bias[n];
    int mbase = (lane >> 4) * 8;
#pragma unroll
    for (int r = 0; r < 8; ++r)
        Y[(mbase + r) * 128 + n] = acc[r] + bn;
}

// ---- Kernel 4: L2-normalize rows, replicate reference diagonal permutation,
//      cross_gate[idx] = sigmoid(C * rgb_n[k][k] * t_n[k][c2]), k=idx%4, c2=idx/4
__global__ __launch_bounds__(512) void k_cross_gate(const float* __restrict__ Y,
                                                    float* __restrict__ cross) {
    __shared__ float norms[8];
    int tid = threadIdx.x;
    if (tid < 8) {
        float s = 0.f;
        for (int c = 0; c < 128; ++c) { float v = Y[tid * 128 + c]; s += v * v; }
        norms[tid] = sqrtf(s);
    }
    __syncthreads();
    int k = tid & 3, c2 = tid >> 2;
    float ry = Y[k * 128 + k] / norms[k];           // rgb rows 0-3
    float ty = Y[(4 + k) * 128 + c2] / norms[4 + k]; // t rows 4-7
    cross[tid] = sigmoidf_((float)Cc * ry * ty);
}

// ---- Kernel 5: fused depthwise-group conv 7x7 + bias + channel max.
//      One block per (b, group g). Group g reads 2 input channels (RGB if g<64
//      else T), gates folded into LDS weights, produces maxima of out channels
//      2g and 2g+1. LDS strip tiling: 8 output rows -> 14x168 padded input rows.
#define LROWS 14
#define LW    168
__global__ __launch_bounds__(256) void k_dwconv_max(const float* __restrict__ RGB,
                                                    const float* __restrict__ T,
                                                    const float* __restrict__ cross,
                                                    const float* __restrict__ dw_w,  // (256,2,7,7)
                                                    const float* __restrict__ dw_b,
                                                    float* __restrict__ dwS) {       // stacked 16x256
    int blk = blockIdx.x;                  // 0..511
    int b = blk >> 7, g = blk & 127;
    const float* src; int ic0;
    if (g < 64) { src = RGB + (size_t)b * Cc * HWp; ic0 = 2 * g; }
    else        { src = T   + (size_t)b * Cc * HWp; ic0 = 2 * g - 128; }

    __shared__ float wsm[2][2 * 49];            // [out sub-channel][in ch * tap]
    __shared__ float tile[2][LROWS][LW];
    __shared__ float red0[256], red1[256];

    int tid = threadIdx.x;
    if (tid < 196) {                            // j*98 + i*49 + k
        int j = tid / 98, rem = tid - j * 98;
        int i = rem / 49, k = rem - i * 49;
        int o = 2 * g + j;
        float gate = cross[b * 128 + ic0 + i];
        wsm[j][i * 49 + k] = dw_w[(o * 2 + i) * 49 + k] * gate;
    }
    float bias0 = dw_b[2 * g], bias1 = dw_b[2 * g + 1];
    float max0 = -3.4e38f, max1 = -3.4e38f;

    for (int s = 0; s < 20; ++s) {              // 20 strips of 8 output rows
        __syncthreads();
        for (int idx = tid; idx < 2 * LROWS * LW; idx += 256) {
            int i = idx / (LROWS * LW);
            int rem = idx - i * (LROWS * LW);
            int r = rem / LW, cc = rem - r * LW;
            int gr = s * 8 - 3 + r, gc = cc - 3;
            float v = 0.f;
            if ((unsigned)gr < (unsigned)Hh && (unsigned)gc < (unsigned)Ww)
                v = src[(size_t)(ic0 + i) * HWp + gr * Ww + gc];
            tile[i][r][cc] = v;
        }
        __syncthreads();
        for (int p = tid; p < 8 * Ww; p += 256) {
            int oh = p / Ww, ow = p - oh * Ww;
            float a0 = bias0, a1 = bias1;
#pragma unroll
            for (int i = 0; i < 2; ++i)
#pragma unroll
                for (int khh = 0; khh < 7; ++khh)
#pragma unroll
                    for (int kww = 0; kww < 7; ++kww) {
                        float v = tile[i][oh + khh][ow + kww];
                        int wi = i * 49 + khh * 7 + kww;
                        a0 += v * wsm[0][wi];
                        a1 += v * wsm[1][wi];
                    }
            max0 = fmaxf(max0, a0);
            max1 = fmaxf(max1, a1);
        }
    }
    red0[tid] = max0; red1[tid] = max1;
    __syncthreads();
    for (int s = 128; s > 0; s >>= 1) {
        if (tid < s) {
            red0[tid] = fmaxf(red0[tid], red0[tid + s]);
            red1[tid] = fmaxf(red1[tid], red1[tid + s]);
        }
        __syncthreads();
    }
    if (tid == 0) {
        dwS[b * 256 + 2 * g]     = red0[0];
        dwS[b * 256 + 2 * g + 1] = red1[0];
    }
}

// ---- Kernel 6: caH(16x16) = gelu( dwS(16x256) @ ca_down_w^T + b ), WMMA, 1 wave
__global__ __launch_bounds__(32) void k_ca_gemm1(const float* __restrict__ dwS,
                                                 const float* __restrict__ W,   // (16,256)
                                                 const float* __restrict__ bias,
                                                 float* __restrict__ caH) {
    const int lane = threadIdx.x & 31;
    const int mr = lane & 15;
    const int kh = (lane >> 4) * 2;
    v8f acc = {};
    for (int k0 = 0; k0 < 256; k0 += 4) {
        v2f a, bm;
        a.x  = dwS[mr * 256 + k0 + kh];
        a.y  = dwS[mr * 256 + k0 + kh + 1];
        bm.x = W[mr * 256 + k0 + kh];          // n == mr here (N=16)
        bm.y = W[mr * 256 + k0 + kh + 1];
        acc = __builtin_amdgcn_wmma_f32_16x16x4_f32(false, a, false, bm,
                                                    (short)0, acc, false, false);
    }
    float bn = bias[mr];
    int mbase = (lane >> 4) * 8;
#pragma unroll
    for (int r = 0; r < 8; ++r)
        caH[(mbase + r) * 16 + mr] = gelu_exact(acc[r] + bn);
}

// ---- Kernel 7: fuse_gate = sigmoid(caH(16x16) @ ca_up_w^T + b) [rows 0-3],
//      then per-(b,c) scale/coef tables for the final fusion pass. WMMA.
__global__ __launch_bounds__(512) void k_ca_gemm2_tables(const float* __restrict__ caH,
                                                         const float* __restrict__ W,  // (256,16)
                                                         const float* __restrict__ bias,
                                                         const float* __restrict__ cross,
                                                         const float* __restrict__ sr_w,
                                                         const float* __restrict__ st_w,
                                                         float* __restrict__ scaleR,
                                                         float* __restrict__ scaleT,
                                                         float* __restrict__ coefR,
                                                         float* __restrict__ coefT) {
    __shared__ float fuse[4 * 256];
    const int tid = threadIdx.x;
    const int lane = tid & 31;
    const int wave = tid >> 5;           // 0..15
    const int n0 = wave * 16;
    const int mr = lane & 15;
    const int kh = (lane >> 4) * 2;
    const int n = n0 + mr;
    v8f acc = {};
    for (int k0 = 0; k0 < 16; k0 += 4) {
        v2f a, bm;
        a.x  = caH[mr * 16 + k0 + kh];
        a.y  = caH[mr * 16 + k0 + kh + 1];
        bm.x = W[n * 16 + k0 + kh];
        bm.y = W[n * 16 + k0 + kh + 1];
        acc = __builtin_amdgcn_wmma_f32_16x16x4_f32(false, a, false, bm,
                                                    (short)0, acc, false, false);
    }
    float bn = bias[n];
    int mbase = (lane >> 4) * 8;
#pragma unroll
    for (int r = 0; r < 8; ++r) {
        int m = mbase + r;
        if (m < 4) fuse[m * 256 + n] = sigmoidf_(acc[r] + bn);
    }
    __syncthreads();
    // tid -> (b,c); scale = g*fuse + (1-g); coef = scale * einsum weight
    int b = tid >> 7, c = tid & 127;
    float g = cross[tid];
    float sr = g * fuse[b * 256 + c]       + 1.0f - g;
    float st = g * fuse[b * 256 + 128 + c] + 1.0f - g;
    scaleR[tid] = sr;         scaleT[tid] = st;
    coefR[tid]  = sr * sr_w[c]; coefT[tid] = st * st_w[c];
}

// ---- Kernel 8: streaming per-pixel fusion. float4 (b128) loads; pass1 = two
//      channel dot products, softmax2 = sigmoid(fr-ft), pass2 rescale+store.
__global__ __launch_bounds__(256) void k_fuse(const float* __restrict__ RGB,
                                              const float* __restrict__ T,
                                              const float* __restrict__ scaleR,
                                              const float* __restrict__ scaleT,
                                              const float* __restrict__ coefR,
                                              const float* __restrict__ coefT,
                                              const float* __restrict__ sr_b,
                                              const float* __restrict__ st_b,
                                              float* __restrict__ outR,
                                              float* __restrict__ outT) {
    int b = blockIdx.y;
    int v = blockIdx.x * 256 + threadIdx.x;     // float4 index 0..6399
    __shared__ float sR[128], sT[128], cR[128], cT[128];
    if (threadIdx.x < 128) {
        int c = threadIdx.x;
        sR[c] = scaleR[b * 128 + c]; sT[c] = scaleT[b * 128 + c];
        cR[c] = coefR[b * 128 + c];  cT[c] = coefT[b * 128 + c];
    }
    __syncthreads();

    const float4* R4 = (const float4*)(RGB + (size_t)b * Cc * HWp);
    const float4* T4 = (const float4*)(T   + (size_t)b * Cc * HWp);
    const int st4 = HWp / 4;

    float srb = sr_b[0], stb = st_b[0];
    float4 fr = make_float4(srb, srb, srb, srb);
    float4 ft = make_float4(stb, stb, stb, stb);
    for (int c = 0; c < 128; ++c) {
        if (c + 8 < 128) {
            __builtin_prefetch(&R4[(size_t)(c + 8) * st4 + v], 0, 1);
            __builtin_prefetch(&T4[(size_t)(c + 8) * st4 + v], 0, 1);
        }
        float4 x = R4[(size_t)c * st4 + v];
        float4 y = T4[(size_t)c * st4 + v];
        float wr = cR[c], wt = cT[c];
        fr.x += x.x * wr; fr.y += x.y * wr; fr.z += x.z * wr; fr.w += x.w * wr;
        ft.x += y.x * wt; ft.y += y.y * wt; ft.z += y.z * wt; ft.w += y.w * wt;
    }
    float4 ar;
    ar.x = sigmoidf_(fr.x - ft.x); ar.y = sigmoidf_(fr.y - ft.y);
    ar.z = sigmoidf_(fr.z - ft.z); ar.w = sigmoidf_(fr.w - ft.w);

    float4* OR4 = (float4*)(outR + (size_t)b * Cc * HWp);
    float4* OT4 = (float4*)(outT + (size_t)b * Cc * HWp);
    for (int c = 0; c < 128; ++c) {
        float4 x = R4[(size_t)c * st4 + v];       // L2-resident re-read
        float4 y = T4[(size_t)c * st4 + v];
        float mr = sR[c], mt = sT[c];
        float4 o1, o2;
        o1.x = x.x * mr * ar.x;          o1.y = x.y * mr * ar.y;
        o1.z = x.z * mr * ar.z;          o1.w = x.w * mr * ar.w;
        o2.x = y.x * mt * (1.f - ar.x);  o2.y = y.y * mt * (1.f - ar.y);
        o2.z = y.z * mt * (1.f - ar.z);  o2.w = y.w * mt * (1.f - ar.w);
        OR4[(size_t)c * st4 + v] = o1;
        OT4[(size_t)c * st4 + v] = o2;
    }
}

// ---------------------------------------------------------------------------
extern "C" void kernel_launch(void* const* d_in, const int* in_sizes, int n_in,
                              void* d_out, int out_size, void* d_ws, size_t ws_size,
                              hipStream_t stream) {
    const float* RGB       = (const float*)d_in[0];
    const float* T         = (const float*)d_in[1];
    const float* fp_down_w = (const float*)d_in[2];
    const float* fp_down_b = (const float*)d_in[3];
    const float* fp_up_w   = (const float*)d_in[4];
    const float* fp_up_b   = (const float*)d_in[5];
    const float* dw_w      = (const float*)d_in[6];
    const float* dw_b      = (const float*)d_in[7];
    const float* ca_down_w = (const float*)d_in[8];
    const float* ca_down_b = (const float*)d_in[9];
    const float* ca_up_w   = (const float*)d_in[10];
    const float* ca_up_b   = (const float*)d_in[11];
    const float* sr_w      = (const float*)d_in[12];
    const float* sr_b      = (const float*)d_in[13];
    const float* st_w      = (const float*)d_in[14];
    const float* st_b      = (const float*)d_in[15];

    float* ws = (float*)d_ws;
    float* Mstack = ws;            // 16*128  = 2048
    float* H1     = ws + 2048;     // 16*256  = 4096
    float* Y      = ws + 6144;     // 16*128  = 2048
    float* cross  = ws + 8192;     // 512
    float* dwS    = ws + 8704;     // 16*256  = 4096 (rows 4-15 stay zero)
    float* caH    = ws + 12800;    // 16*16   = 256
    float* scaleR = ws + 13056;    // 512
    float* scaleT = ws + 13568;    // 512
    float* coefR  = ws + 14080;    // 512
    float* coefT  = ws + 14592;    // 512  -> total 15104 floats (~60 KB)

    hipMemsetAsync(d_ws, 0, 15104 * sizeof(float), stream);

    float* outR = (float*)d_out;
    float* outT = outR + (size_t)Bb * Cc * HWp;

    k_pool_mean<<<2 * Bb * Cc, 256, 0, stream>>>(RGB, T, Mstack);
    k_fp_gemm1<<<1, 512, 0, stream>>>(Mstack, fp_down_w, fp_down_b, H1);
    k_fp_gemm2<<<1, 256, 0, stream>>>(H1, fp_up_w, fp_up_b, Y);
    k_cross_gate<<<1, 512, 0, stream>>>(Y, cross);
    k_dwconv_max<<<Bb * Cc, 256, 0, stream>>>(RGB, T, cross, dw_w, dw_b, dwS);
    k_ca_gemm1<<<1, 32, 0, stream>>>(dwS, ca_down_w, ca_down_b, caH);
    k_ca_gemm2_tables<<<1, 512, 0, stream>>>(caH, ca_up_w, ca_up_b, cross,
                                             sr_w, st_w, scaleR, scaleT, coefR, coefT);
    dim3 grid((HWp / 4) / 256, Bb);   // 25 x 4
    k_fuse<<<grid, 256, 0, stream>>>(RGB, T, scaleR, scaleT, coefR, coefT,
                                     sr_b, st_b, outR, outT);
}